// MaskedScaleDotProductAttention_31765578121671
// MI455X (gfx1250) — compile-verified
//
#include <hip/hip_runtime.h>

#define D_IN   512
#define D_OUT  512
#define D_K    64
#define NB     4
#define NSEQ   2048
#define LN_EPS 1e-6f
#define NEG_BIG (-1e9f)

typedef __attribute__((ext_vector_type(2))) float v2f;
typedef __attribute__((ext_vector_type(8))) float v8f;
typedef unsigned int u32x4 __attribute__((ext_vector_type(4)));
typedef int i32x4 __attribute__((ext_vector_type(4)));
typedef int i32x8 __attribute__((ext_vector_type(8)));

// D = A(16x4, f32) * B(4x16, f32) + C(16x16, f32)   -- full fp32 precision
// A frag per lane: lanes 0-15 hold row m=l16, K = k0+{0,1}; lanes 16-31: K = k0+{2,3}
// B frag per lane: lanes 0-15 hold col n=l16, K rows k0+{0,1}; lanes 16-31: k0+{2,3}
// C/D: VGPR r -> row (r + 8*half), col = l16
__device__ __forceinline__ v8f wmma_f32(v2f a, v2f b, v8f c) {
  return __builtin_amdgcn_wmma_f32_16x16x4_f32(
      /*neg_a=*/false, a, /*neg_b=*/false, b,
      /*c_mod=*/(short)0, c, /*reuse_a=*/false, /*reuse_b=*/false);
}

// ---------------------------------------------------------------------------
// Tensor Data Mover: 2D tile load Global -> LDS (D# per CDNA5 ISA)
//   g0: count=1 | lds_addr | global_addr(bytes) | type=2
//   g1: w0flags(data_size/pad) | tensor_dim0/1 | tile_dim0/1 | dim0_stride
//   dims & strides in data_size(=4B) element units.
// This toolchain exposes the 6-arg builtin:
//   (u32x4 g0, i32x8 g1, i32x4 g2, i32x4 g3, i32x8 extra, i32 cpol)
// ---------------------------------------------------------------------------
__device__ __forceinline__ void tdm_load_2d(unsigned lds_addr,
                                            unsigned long long gaddr_bytes,
                                            unsigned tensor_d0, unsigned tensor_d1,
                                            unsigned tile_d0, unsigned tile_d1,
                                            unsigned long long stride0,
                                            unsigned w0flags) {
  u32x4 g0;
  g0[0] = 1u;                                             // count=1, user mode
  g0[1] = lds_addr;                                       // LDS byte address
  g0[2] = (unsigned)gaddr_bytes;                          // global addr [31:0]
  g0[3] = (unsigned)((gaddr_bytes >> 32) & 0x01FFFFFFull) // global addr [56:32]
          | (2u << 30);                                   // type = 2 ("image")
  i32x8 g1;
  g1[0] = (int)w0flags;                                   // data_size=4B | pad cfg
  g1[1] = (int)((tensor_d0 & 0xFFFFu) << 16);             // tensor_dim0[15:0]
  g1[2] = (int)((tensor_d0 >> 16) | ((tensor_d1 & 0xFFFFu) << 16));
  g1[3] = (int)((tensor_d1 >> 16) | ((tile_d0 & 0xFFFFu) << 16));
  g1[4] = (int)(tile_d1 & 0xFFFFu);                       // tile_dim1; tile_dim2=0
  g1[5] = (int)(unsigned)stride0;                         // dim0_stride[31:0]
  g1[6] = (int)((unsigned)(stride0 >> 32) & 0xFFFFu);     // dim0_stride[47:32]
  g1[7] = 0;                                              // dim1_stride unused (2D)
  i32x4 z4 = {0, 0, 0, 0};
  i32x8 z8 = {0, 0, 0, 0, 0, 0, 0, 0};
  __builtin_amdgcn_tensor_load_to_lds(g0, g1, z4, z4, z8, 0);
}

// data_size = 4B (enc 2) at bits[17:16]; pad_enable bit20; pad_interval bits[24:22];
// pad_amount bits[31:25].
// V slab: pad 8 dwords every 256 dwords  -> LDS row stride 528 floats
#define TDM_FLAGS_V ((2u << 16) | (1u << 20) | (7u << 22) | (7u << 25))
// A tile: pad 2 dwords every 32 dwords   -> LDS row stride 34 floats
#define TDM_FLAGS_A ((2u << 16) | (1u << 20) | (4u << 22) | (1u << 25))

#define KT 32            // K-slab depth staged per iteration
#define SV_STRIDE 528    // 512 + 2*8 pad
#define SA_STRIDE 34     // 32 + 2 pad

// ---------------------------------------------------------------------------
// Kernel 1: Y[8192 x 64] = X[8192 x 512] * W[512 x 64], result scaled.
// One wave computes a 16x64 output tile (4 WMMA column sub-tiles).
// ---------------------------------------------------------------------------
__global__ __launch_bounds__(32) void k_proj(const float* __restrict__ X,
                                             const float* __restrict__ W,
                                             float* __restrict__ Y,
                                             float scale) {
  const int lane = threadIdx.x & 31;
  const int half = lane >> 4;
  const int l16  = lane & 15;
  const int m0   = blockIdx.x * 16;

  v8f acc[4] = {};
  const float* Arow = X + (size_t)(m0 + l16) * D_IN;

  for (int k0 = 0; k0 < D_IN; k0 += 4) {
    v2f a;
    a.x = Arow[k0 + 2 * half];
    a.y = Arow[k0 + 2 * half + 1];
#pragma unroll
    for (int s = 0; s < 4; ++s) {
      const int n = s * 16 + l16;
      v2f b;
      b.x = W[(size_t)(k0 + 2 * half) * D_K + n];
      b.y = W[(size_t)(k0 + 2 * half + 1) * D_K + n];
      acc[s] = wmma_f32(a, b, acc[s]);
    }
  }

#pragma unroll
  for (int s = 0; s < 4; ++s)
#pragma unroll
    for (int r = 0; r < 8; ++r) {
      const int row = m0 + r + 8 * half;
      const int col = s * 16 + l16;
      Y[(size_t)row * D_K + col] = acc[s][r] * scale;
    }
}

// ---------------------------------------------------------------------------
// Kernel 2: scores = qp * kp^T (qp pre-scaled), mask applied, raw scores
// written into the attn output region. One wave per 16x64 score tile, K=64.
// ---------------------------------------------------------------------------
__global__ __launch_bounds__(32) void k_scores(const float* __restrict__ qp,
                                               const float* __restrict__ kp,
                                               const int* __restrict__ mask,
                                               float* __restrict__ attn) {
  const int lane = threadIdx.x & 31;
  const int half = lane >> 4;
  const int l16  = lane & 15;
  const int m0     = blockIdx.x * 16;
  const int n0base = blockIdx.y * 64;
  const int b      = blockIdx.z;

  const float* Q = qp + (size_t)b * NSEQ * D_K;
  const float* K = kp + (size_t)b * NSEQ * D_K;

  v2f afr[16];
  const float* Arow = Q + (size_t)(m0 + l16) * D_K;
#pragma unroll
  for (int kk = 0; kk < 16; ++kk) {
    afr[kk].x = Arow[4 * kk + 2 * half];
    afr[kk].y = Arow[4 * kk + 2 * half + 1];
  }

  v8f acc[4];
#pragma unroll
  for (int s = 0; s < 4; ++s) {
    const float* Bcol = K + (size_t)(n0base + s * 16 + l16) * D_K;
    v8f a = {};
#pragma unroll
    for (int kk = 0; kk < 16; ++kk) {
      v2f bf;
      bf.x = Bcol[4 * kk + 2 * half];
      bf.y = Bcol[4 * kk + 2 * half + 1];
      a = wmma_f32(afr[kk], bf, a);
    }
    acc[s] = a;
  }

#pragma unroll
  for (int s = 0; s < 4; ++s)
#pragma unroll
    for (int r = 0; r < 8; ++r) {
      const int row = m0 + r + 8 * half;
      const int col = n0base + s * 16 + l16;
      const size_t idx = ((size_t)b * NSEQ + row) * NSEQ + col;
      attn[idx] = (mask[idx] == 0) ? NEG_BIG : acc[s][r];
    }
}

// ---------------------------------------------------------------------------
// Kernel 3: in-place row softmax over 2048 columns. One 256-thread block per
// row; each thread holds 8 values in registers (single read + single write).
// ---------------------------------------------------------------------------
__global__ __launch_bounds__(256) void k_softmax(float* __restrict__ attn) {
  __shared__ float red[256];
  const int tid = threadIdx.x;
  float* p = attn + (size_t)blockIdx.x * NSEQ;

  float vals[8];
#pragma unroll
  for (int j = 0; j < 8; ++j) vals[j] = p[tid + j * 256];

  float m = vals[0];
#pragma unroll
  for (int j = 1; j < 8; ++j) m = fmaxf(m, vals[j]);
  red[tid] = m;
  __syncthreads();
  for (int off = 128; off > 0; off >>= 1) {
    if (tid < off) red[tid] = fmaxf(red[tid], red[tid + off]);
    __syncthreads();
  }
  m = red[0];
  __syncthreads();

  float s = 0.0f;
#pragma unroll
  for (int j = 0; j < 8; ++j) {
    vals[j] = __expf(vals[j] - m);
    s += vals[j];
  }
  red[tid] = s;
  __syncthreads();
  for (int off = 128; off > 0; off >>= 1) {
    if (tid < off) red[tid] += red[tid + off];
    __syncthreads();
  }
  const float inv = 1.0f / red[0];

#pragma unroll
  for (int j = 0; j < 8; ++j) p[tid + j * 256] = vals[j] * inv;
}

// ---------------------------------------------------------------------------
// Kernel 4: out1 = attn @ v + v.
// One 256-thread block (8 waves) per 16-row output stripe (16 x 512).
// K-slabs of attn (16 x 32) and V (32 x 512) are staged into LDS by the
// Tensor Data Mover (TENSOR_LOAD_TO_LDS) with TDM padding configured to make
// both the A-fragment and B-fragment WMMA read patterns bank-conflict-free.
// ---------------------------------------------------------------------------
__global__ __launch_bounds__(256) void k_attnv(const float* __restrict__ attn,
                                               const float* __restrict__ v,
                                               float* __restrict__ out1) {
  __shared__ float sV[KT * SV_STRIDE];       // 32 x 528 floats (~66 KB)
  __shared__ float sA[16 * SA_STRIDE];       // 16 x 34 floats

  const int tid  = threadIdx.x;
  const int wave = tid >> 5;
  const int lane = tid & 31;
  const int half = lane >> 4;
  const int l16  = lane & 15;
  const int m0   = blockIdx.x * 16;
  const int b    = blockIdx.y;

  const float* A = attn + ((size_t)b * NSEQ + m0) * NSEQ;   // 16 rows of attn
  const float* V = v + (size_t)b * NSEQ * D_IN;

  const unsigned ldsV = (unsigned)(uintptr_t)(&sV[0]);      // LDS byte address
  const unsigned ldsA = (unsigned)(uintptr_t)(&sA[0]);
  const unsigned long long Vbytes = (unsigned long long)(uintptr_t)V;
  const unsigned long long Abytes = (unsigned long long)(uintptr_t)A;

  // Padded LDS column index for the B fragment (pad 8 dwords every 256)
  int cpad[4];
#pragma unroll
  for (int s = 0; s < 4; ++s) {
    const int col = wave * 64 + s * 16 + l16;
    cpad[s] = col + ((col >> 8) << 3);
  }

  v8f acc[4] = {};

  for (int kt = 0; kt < NSEQ; kt += KT) {
    if (wave == 0) {
      // V slab: KT rows x 512 cols, row-major, stride 512 elements
      tdm_load_2d(ldsV, Vbytes + (unsigned long long)kt * D_IN * 4u,
                  D_IN, NSEQ, D_IN, KT, D_IN, TDM_FLAGS_V);
      // A tile: 16 rows x KT cols out of the 16 x 2048 attn stripe
      tdm_load_2d(ldsA, Abytes + (unsigned long long)kt * 4u,
                  NSEQ, NSEQ, KT, 16, NSEQ, TDM_FLAGS_A);
      __builtin_amdgcn_s_wait_tensorcnt(0);
    }
    __syncthreads();

#pragma unroll
    for (int kk = 0; kk < KT; kk += 4) {
      v2f a;
      a.x = sA[l16 * SA_STRIDE + kk + 2 * half];
      a.y = sA[l16 * SA_STRIDE + kk + 2 * half + 1];
      const int r0 = (kk + 2 * half) * SV_STRIDE;
      const int r1 = (kk + 2 * half + 1) * SV_STRIDE;
#pragma unroll
      for (int s = 0; s < 4; ++s) {
        v2f bf;
        bf.x = sV[r0 + cpad[s]];
        bf.y = sV[r1 + cpad[s]];
        acc[s] = wmma_f32(a, bf, acc[s]);
      }
    }
    __syncthreads();
  }

#pragma unroll
  for (int s = 0; s < 4; ++s)
#pragma unroll
    for (int r = 0; r < 8; ++r) {
      const int row = m0 + r + 8 * half;
      const int col = wave * 64 + s * 16 + l16;
      out1[((size_t)b * NSEQ + row) * D_IN + col] =
          acc[s][r] + V[(size_t)row * D_IN + col];
    }
}

// ---------------------------------------------------------------------------
// Kernel 5: out = LayerNorm(out1 @ w_fc). One 256-thread block (8 waves) per
// 16-row tile; A tile staged in LDS, LDS reused for fc result, then LN.
// ---------------------------------------------------------------------------
__global__ __launch_bounds__(256) void k_fc_ln(const float* __restrict__ out1,
                                               const float* __restrict__ wfc,
                                               const float* __restrict__ ln_g,
                                               const float* __restrict__ ln_b,
                                               float* __restrict__ out) {
  __shared__ float tile[16 * D_OUT];   // 32 KB: A tile, then result tile
  __shared__ float red[16][17];
  __shared__ float mu_s[16], rs_s[16];

  const int tid  = threadIdx.x;
  const int wave = tid >> 5;
  const int lane = tid & 31;
  const int half = lane >> 4;
  const int l16  = lane & 15;
  const int row0 = blockIdx.x * 16;

  // stage A tile (16 x 512)
  for (int i = tid; i < 16 * D_IN; i += 256)
    tile[i] = out1[(size_t)row0 * D_IN + i];
  __syncthreads();

  const int n0base = wave * 64;
  v8f acc[4] = {};
  for (int k0 = 0; k0 < D_IN; k0 += 4) {
    v2f a;
    a.x = tile[l16 * D_IN + k0 + 2 * half];
    a.y = tile[l16 * D_IN + k0 + 2 * half + 1];
#pragma unroll
    for (int s = 0; s < 4; ++s) {
      const int col = n0base + s * 16 + l16;
      v2f bf;
      bf.x = wfc[(size_t)(k0 + 2 * half) * D_OUT + col];
      bf.y = wfc[(size_t)(k0 + 2 * half + 1) * D_OUT + col];
      acc[s] = wmma_f32(a, bf, acc[s]);
    }
  }
  __syncthreads();   // everyone done reading A from LDS

#pragma unroll
  for (int s = 0; s < 4; ++s)
#pragma unroll
    for (int r = 0; r < 8; ++r)
      tile[(r + 8 * half) * D_OUT + n0base + s * 16 + l16] = acc[s][r];
  __syncthreads();

  // LayerNorm: 16 threads per row
  const int gR = tid >> 4;
  const int gj = tid & 15;

  float sum = 0.0f;
  for (int c = gj; c < D_OUT; c += 16) sum += tile[gR * D_OUT + c];
  red[gR][gj] = sum;
  __syncthreads();
  if (gj == 0) {
    float t = 0.0f;
    for (int j = 0; j < 16; ++j) t += red[gR][j];
    mu_s[gR] = t * (1.0f / (float)D_OUT);
  }
  __syncthreads();
  const float mu = mu_s[gR];

  float vs = 0.0f;
  for (int c = gj; c < D_OUT; c += 16) {
    const float d = tile[gR * D_OUT + c] - mu;
    vs += d * d;
  }
  red[gR][gj] = vs;
  __syncthreads();
  if (gj == 0) {
    float t = 0.0f;
    for (int j = 0; j < 16; ++j) t += red[gR][j];
    rs_s[gR] = rsqrtf(t * (1.0f / (float)D_OUT) + LN_EPS);
  }
  __syncthreads();
  const float rstd = rs_s[gR];

  for (int c = gj; c < D_OUT; c += 16) {
    const float val = (tile[gR * D_OUT + c] - mu) * rstd * ln_g[c] + ln_b[c];
    out[(size_t)(row0 + gR) * D_OUT + c] = val;
  }
}

// ---------------------------------------------------------------------------
extern "C" void kernel_launch(void* const* d_in, const int* in_sizes, int n_in,
                              void* d_out, int out_size, void* d_ws, size_t ws_size,
                              hipStream_t stream) {
  const float* q    = (const float*)d_in[0];
  const float* k    = (const float*)d_in[1];
  const float* v    = (const float*)d_in[2];
  const int*   mask = (const int*)d_in[3];
  const float* w_qs = (const float*)d_in[4];
  const float* w_ks = (const float*)d_in[5];
  const float* w_fc = (const float*)d_in[6];
  const float* ln_g = (const float*)d_in[7];
  const float* ln_b = (const float*)d_in[8];

  float* outp = (float*)d_out;                          // [B,N,512]
  float* attn = outp + (size_t)NB * NSEQ * D_OUT;       // [B,N,N]

  float* qp   = (float*)d_ws;                           // [B,N,64]
  float* kp   = qp + (size_t)NB * NSEQ * D_K;           // [B,N,64]
  float* out1 = kp + (size_t)NB * NSEQ * D_K;           // [B,N,512]

  const float scale = 0.125f;  // 1/sqrt(64)

  k_proj<<<dim3(NB * NSEQ / 16), dim3(32), 0, stream>>>(q, w_qs, qp, scale);
  k_proj<<<dim3(NB * NSEQ / 16), dim3(32), 0, stream>>>(k, w_ks, kp, 1.0f);
  k_scores<<<dim3(NSEQ / 16, NSEQ / 64, NB), dim3(32), 0, stream>>>(qp, kp, mask, attn);
  k_softmax<<<dim3(NB * NSEQ), dim3(256), 0, stream>>>(attn);
  k_attnv<<<dim3(NSEQ / 16, NB), dim3(256), 0, stream>>>(attn, v, out1);
  k_fc_ln<<<dim3(NB * NSEQ / 16), dim3(256), 0, stream>>>(out1, w_fc, ln_g, ln_b, outp);
}